// Self_Attention_2199023255569
// MI455X (gfx1250) — compile-verified
//
#include <hip/hip_runtime.h>
#include <hip/hip_bf16.h>
#include <math.h>

// Problem constants (from reference)
constexpr int Bb = 4;
constexpr int Cc = 256;
constexpr int Nn = 4096;   // H*W = 64*64
constexpr int C8 = 32;     // Cc/8

typedef __attribute__((ext_vector_type(2))) float v2f;
typedef __attribute__((ext_vector_type(8))) float v8f;

// ---------------------------------------------------------------------------
// Kernel 1: 1x1-conv projections  Y[d,n] = sum_c W[d,c] * x[b,c,n] + bias[d]
// transposeOut=1 -> store Y[b][n][d] (for q / k^T, ld = D)
// transposeOut=0 -> store Y[b][d][n] (for v, ld = N)
// grid (Nn/16, D/16, Bb), block (16,16). Projections are <7% of total FLOPs.
// ---------------------------------------------------------------------------
__global__ void proj16(const float* __restrict__ x, const float* __restrict__ Wm,
                       const float* __restrict__ bias, float* __restrict__ Y,
                       int D, int transposeOut) {
  __shared__ float Ws[16][17];
  __shared__ float Xs[16][17];
  const int tx = threadIdx.x, ty = threadIdx.y;
  const int b = blockIdx.z;
  const int n = blockIdx.x * 16 + tx;
  const int d = blockIdx.y * 16 + ty;
  const float* xb = x + (size_t)b * Cc * Nn;
  float acc = 0.0f;
  for (int c0 = 0; c0 < Cc; c0 += 16) {
    Ws[ty][tx] = Wm[(size_t)d * Cc + c0 + tx];
    Xs[ty][tx] = xb[(size_t)(c0 + ty) * Nn + n];
    __syncthreads();
#pragma unroll
    for (int cc = 0; cc < 16; ++cc) acc += Ws[ty][cc] * Xs[cc][tx];
    __syncthreads();
  }
  acc += bias[d];
  if (transposeOut)
    Y[((size_t)b * Nn + n) * D + d] = acc;
  else
    Y[((size_t)b * D + d) * Nn + n] = acc;
}

// ---------------------------------------------------------------------------
// Kernel 2: S = q k  (S[n,m] = sum_d q[n,d] * kT[m,d]),   written TRANSPOSED
// into the attention-map output region:  A[b][m][n] = S[n,m]  (raw logits).
// grid (Nn/128, Nn/16, Bb), block 256 = 8 waves; each wave one 16x16 tile,
// K = C8 = 32 -> 8x V_WMMA_F32_16X16X4_F32. LDS transpose buffer turns the
// column-major D-fragment into coalesced 512B row writes.
// ---------------------------------------------------------------------------
__global__ void s_kernel(const float* __restrict__ q, const float* __restrict__ kT,
                         float* __restrict__ A) {
  const int t = threadIdx.x;
  const int w = t >> 5;
  const int lane = t & 31;
  const int b = blockIdx.z;
  const int n0 = blockIdx.x * 128 + w * 16;
  const int m0 = blockIdx.y * 16;

  const int row = lane & 15;          // M (or N) index of fragment
  const int kp  = (lane >> 4) * 2;    // K base for this half-wave

  const float* qrow = q  + ((size_t)b * Nn + (n0 + row)) * C8 + kp;
  const float* krow = kT + ((size_t)b * Nn + (m0 + row)) * C8 + kp;

  v8f acc = {};
#pragma unroll
  for (int kk = 0; kk < C8; kk += 4) {
    v2f af = *(const v2f*)(qrow + kk);
    v2f bf = *(const v2f*)(krow + kk);
    acc = __builtin_amdgcn_wmma_f32_16x16x4_f32(false, af, false, bf,
                                                (short)0, acc, false, false);
  }

  // stage: sT[m_local][n_local]  (pad 130 keeps b32 writes conflict-light)
  __shared__ float sT[16][130];
  const int nhalf = 8 * (lane >> 4);
#pragma unroll
  for (int r = 0; r < 8; ++r)
    sT[row][w * 16 + r + nhalf] = acc[r];
  __syncthreads();

  // coalesced write: 16 m-rows x 128 contiguous n each
  const int mrow = t >> 4;   // 0..15
  const int nb   = t & 15;
  float* Arow = A + (size_t)b * Nn * Nn + (size_t)(m0 + mrow) * Nn
                  + (size_t)blockIdx.x * 128;
#pragma unroll
  for (int i = 0; i < 8; ++i)
    Arow[nb + 16 * i] = sT[mrow][nb + 16 * i];
}

// ---------------------------------------------------------------------------
// Kernel 3: in-place row softmax of A[b][m][:] (== softmax over n of S[:,m],
// exactly the reference's softmax(dim=-2)). One block per (b,m) row:
// single read + single write of the 268MB map, values kept in registers.
// ---------------------------------------------------------------------------
__global__ void softmax_rows(float* __restrict__ A) {
  const int t = threadIdx.x;
  float* row = A + (size_t)blockIdx.x * Nn;

  float vals[16];
  float lmax = -3.402823466e+38f;
#pragma unroll
  for (int i = 0; i < 16; ++i) {
    vals[i] = row[t + 256 * i];
    lmax = fmaxf(lmax, vals[i]);
  }
  __shared__ float red[256];
  red[t] = lmax;
  __syncthreads();
  for (int s = 128; s > 0; s >>= 1) {
    if (t < s) red[t] = fmaxf(red[t], red[t + s]);
    __syncthreads();
  }
  const float M = red[0];
  __syncthreads();

  float lsum = 0.0f;
#pragma unroll
  for (int i = 0; i < 16; ++i) {
    vals[i] = __expf(vals[i] - M);
    lsum += vals[i];
  }
  red[t] = lsum;
  __syncthreads();
  for (int s = 128; s > 0; s >>= 1) {
    if (t < s) red[t] += red[t + s];
    __syncthreads();
  }
  const float inv = 1.0f / red[0];
#pragma unroll
  for (int i = 0; i < 16; ++i)
    row[t + 256 * i] = vals[i] * inv;
}

// ---------------------------------------------------------------------------
// Kernel 4: o[c,m] = sum_n v[c,n] * A[m,n]  (NT GEMM, K=4096), fused epilogue
// out = x + gamma*o. grid (Nn/16, 1, Bb), block 256 = 8 waves.
// One block covers the FULL 256c x 16m column block so the 268MB attention
// map is read exactly once. Each wave owns TWO 16x16 accumulators (c, c+16)
// sharing the B-fragment -> 2048 fp32 WMMAs/wave, half the B-frag LDS reads.
// K-chunks of 32 staged in LDS (pad 34 keeps 8B-aligned b64 LDS loads);
// __builtin_prefetch pulls the next K-chunk (global_prefetch_b8) while the
// current one is consumed.
// ---------------------------------------------------------------------------
constexpr int KT  = 32;
constexpr int KTP = 34;

__global__ void o_epilogue(const float* __restrict__ v, const float* __restrict__ A,
                           const float* __restrict__ x, const float* __restrict__ gamma,
                           float* __restrict__ out) {
  __shared__ float vS[256][KTP];   // 34,816 B
  __shared__ float aS[16][KTP];    //  2,176 B
  const int t = threadIdx.x;
  const int w = t >> 5;
  const int lane = t & 31;
  const int b = blockIdx.z;
  const int m0 = blockIdx.x * 16;

  const float* vb = v + (size_t)b * Cc * Nn;
  const float* Ab = A + (size_t)b * Nn * Nn;

  const int row = lane & 15;
  const int kp  = (lane >> 4) * 2;

  v8f acc0 = {};
  v8f acc1 = {};
  for (int k0 = 0; k0 < Nn; k0 += KT) {
    // cooperative, coalesced LDS fills: one wave fills one 32-float row/pass
#pragma unroll
    for (int i = 0; i < 32; ++i) {
      const int idx = t + 256 * i;
      vS[idx >> 5][idx & 31] = vb[(size_t)(idx >> 5) * Nn + k0 + (idx & 31)];
    }
#pragma unroll
    for (int i = 0; i < 2; ++i) {
      const int idx = t + 256 * i;
      aS[idx >> 5][idx & 31] = Ab[(size_t)(m0 + (idx >> 5)) * Nn + k0 + (idx & 31)];
    }
    // prefetch next chunk: 32 floats = one 128B line; 256 threads cover all
    // 256 v-rows, threads 0..15 additionally cover the 16 A-rows.
    const int k1 = k0 + KT;
    if (k1 < Nn) {
      __builtin_prefetch(&vb[(size_t)t * Nn + k1], 0, 3);
      if (t < 16) __builtin_prefetch(&Ab[(size_t)(m0 + t) * Nn + k1], 0, 3);
    }
    __syncthreads();
#pragma unroll
    for (int kk = 0; kk < KT; kk += 4) {
      v2f bf  = *(const v2f*)&aS[row][kk + kp];
      v2f af0 = *(const v2f*)&vS[w * 32 + row][kk + kp];
      v2f af1 = *(const v2f*)&vS[w * 32 + 16 + row][kk + kp];
      acc0 = __builtin_amdgcn_wmma_f32_16x16x4_f32(false, af0, false, bf,
                                                   (short)0, acc0, false, false);
      acc1 = __builtin_amdgcn_wmma_f32_16x16x4_f32(false, af1, false, bf,
                                                   (short)0, acc1, false, false);
    }
    __syncthreads();
  }

  const float g = gamma[0];
  const int m = m0 + row;
  const int chalf = 8 * (lane >> 4);
#pragma unroll
  for (int r = 0; r < 8; ++r) {
    const int c0 = w * 32 + r + chalf;          // tile 0: c in [w*32, w*32+16)
    const size_t i0 = ((size_t)b * Cc + c0) * (size_t)Nn + m;
    out[i0] = x[i0] + g * acc0[r];
    const int c1 = c0 + 16;                     // tile 1: c in [w*32+16, w*32+32)
    const size_t i1 = ((size_t)b * Cc + c1) * (size_t)Nn + m;
    out[i1] = x[i1] + g * acc1[r];
  }
}

// ---------------------------------------------------------------------------
// Launch: inputs (x, Wq, bq, Wk, bk, Wv, bv, gamma) all fp32.
// d_out = [out (B*C*N) | attention_map (B*N*N)]. Workspace: q,kT ([B][N][32])
// and v ([B][C][N]) = 20 MB.
// ---------------------------------------------------------------------------
extern "C" void kernel_launch(void* const* d_in, const int* in_sizes, int n_in,
                              void* d_out, int out_size, void* d_ws, size_t ws_size,
                              hipStream_t stream) {
  const float* x     = (const float*)d_in[0];
  const float* Wq    = (const float*)d_in[1];
  const float* bq    = (const float*)d_in[2];
  const float* Wk    = (const float*)d_in[3];
  const float* bk    = (const float*)d_in[4];
  const float* Wv    = (const float*)d_in[5];
  const float* bv    = (const float*)d_in[6];
  const float* gamma = (const float*)d_in[7];

  float* out  = (float*)d_out;
  float* Amap = out + (size_t)Bb * Cc * Nn;   // attention-map region doubles as S scratch

  float* q  = (float*)d_ws;                    // [B][N][32]
  float* kT = q  + (size_t)Bb * Nn * C8;       // [B][N][32] (k stored transposed)
  float* v  = kT + (size_t)Bb * Nn * C8;       // [B][C][N]

  dim3 blk16(16, 16);
  proj16<<<dim3(Nn / 16, C8 / 16, Bb), blk16, 0, stream>>>(x, Wq, bq, q,  C8, 1);
  proj16<<<dim3(Nn / 16, C8 / 16, Bb), blk16, 0, stream>>>(x, Wk, bk, kT, C8, 1);
  proj16<<<dim3(Nn / 16, Cc / 16, Bb), blk16, 0, stream>>>(x, Wv, bv, v,  Cc, 0);

  s_kernel<<<dim3(Nn / 128, Nn / 16, Bb), 256, 0, stream>>>(q, kT, Amap);
  softmax_rows<<<dim3(Bb * Nn), 256, 0, stream>>>(Amap);
  o_epilogue<<<dim3(Nn / 16, 1, Bb), 256, 0, stream>>>(v, Amap, x, gamma, out);
}